// dsq_90013924589944
// MI455X (gfx1250) — compile-verified
//
#include <hip/hip_runtime.h>
#include <math.h>

// DSQ (single-range, n=256 levels, l=-1) elementwise quantizer for MI455X.
//
// Roofline: 67.1M fp32 elements, 537 MB strictly-streaming traffic vs
// 23.3 TB/s HBM => ~23 us floor. Per element: ~9 VALU + 1 v_tanh_f32
// (TRANS32, co-executes with VALU), far below what 65k wave32s can issue
// against that stream. Pure data-movement kernel: b128 NT loads/stores,
// 2 loads in flight per loop iteration, grid-stride.

typedef __attribute__((ext_vector_type(4))) float v4f;

__device__ __forceinline__ float dsq_tanh(float x) {
#if __has_builtin(__builtin_amdgcn_tanhf)
    return __builtin_amdgcn_tanhf(x);     // v_tanh_f32
#else
    float r;
    asm("v_tanh_f32 %0, %1" : "=v"(r) : "v"(x));
    return r;
#endif
}

__device__ __forceinline__ float frcp(float x) {
    return __builtin_amdgcn_rcpf(x);      // v_rcp_f32 (uniform-scalar setup only)
}

struct DsqConsts {
    float l, u, inv_delta, b0, scale, a2, a3, c0, c1, c2;
};

__device__ __forceinline__ DsqConsts make_consts(float k, float u) {
    constexpr float l   = -1.0f;
    constexpr float nm1 = 255.0f;
    DsqConsts c;
    c.l = l;
    c.u = u;
    const float inv_span = frcp(u - l);            // 1/(u-l)
    const float delta    = (u - l) * (1.0f / nm1);
    c.inv_delta          = nm1 * inv_span;
    const float S0       = __logf(fmaf(2.0f, frcp(k), -1.0f));  // log(2/k - 1)
    c.scale              = S0 * c.inv_delta;
    c.b0                 = -l * c.inv_delta - 0.5f;
    c.a2                 = -S0;
    c.a3                 = -c.scale * l - 0.5f * S0;
    c.c0                 = l + 0.5f * delta;
    c.c1                 = delta;
    c.c2                 = 0.5f * delta * frcp(1.0f - k);
    return c;
}

__device__ __forceinline__ float dsq_one(float xx, const DsqConsts& c) {
    float xc  = fminf(fmaxf(xx, c.l), c.u);                               // clip
    float ii  = fminf(fmaxf(ceilf(fmaf(xx, c.inv_delta, c.b0)), 0.0f),
                      255.0f);                                            // bin idx
    float arg = fmaf(c.scale, xc, fmaf(c.a2, ii, c.a3));                  // scale*(xc-mi)
    return fmaf(c.c1, ii, fmaf(c.c2, dsq_tanh(arg), c.c0));               // x_hat
}

__global__ void __launch_bounds__(256)
dsq_kernel(const float* __restrict__ x,
           const float* __restrict__ kp,
           const float* __restrict__ up,
           float* __restrict__ out,
           long long nvec,     // number of float4 groups
           long long n)        // total element count
{
    // Uniform scalars (s_load path), setup uses v_rcp/v_log only.
    const DsqConsts c = make_consts(kp[0], up[0]);

    const v4f* __restrict__ xv = (const v4f*)x;
    v4f* __restrict__ ov       = (v4f*)out;

    const long long stride = (long long)gridDim.x * blockDim.x;
    long long i0 = (long long)blockIdx.x * blockDim.x + threadIdx.x;
    long long i1 = i0 + stride;

    // Main loop: two b128 NT loads in flight, then compute, then two NT stores.
    for (; i1 < nvec; i0 += 2 * stride, i1 += 2 * stride) {
        v4f v0 = __builtin_nontemporal_load(&xv[i0]);
        v4f v1 = __builtin_nontemporal_load(&xv[i1]);
        v4f r0, r1;
#pragma unroll
        for (int j = 0; j < 4; ++j) r0[j] = dsq_one(v0[j], c);
#pragma unroll
        for (int j = 0; j < 4; ++j) r1[j] = dsq_one(v1[j], c);
        __builtin_nontemporal_store(r0, &ov[i0]);
        __builtin_nontemporal_store(r1, &ov[i1]);
    }
    // Odd leftover float4.
    if (i0 < nvec) {
        v4f v0 = __builtin_nontemporal_load(&xv[i0]);
        v4f r0;
#pragma unroll
        for (int j = 0; j < 4; ++j) r0[j] = dsq_one(v0[j], c);
        __builtin_nontemporal_store(r0, &ov[i0]);
    }

    // Scalar tail (n not divisible by 4); never runs for the reference shape.
    for (long long t = 4 * nvec + ((long long)blockIdx.x * blockDim.x + threadIdx.x);
         t < n; t += stride) {
        out[t] = dsq_one(x[t], c);
    }
}

extern "C" void kernel_launch(void* const* d_in, const int* in_sizes, int n_in,
                              void* d_out, int out_size, void* d_ws, size_t ws_size,
                              hipStream_t stream) {
    // setup_inputs() order: xx (f32 [16384,4096]), k (f32 [1]), u (f32 [1])
    const float* x  = (const float*)d_in[0];
    const float* kp = (const float*)d_in[1];
    const float* up = (const float*)d_in[2];
    float* out      = (float*)d_out;

    const long long n    = (long long)out_size;
    const long long nvec = n >> 2;

    const int threads = 256;                      // 8 wave32s per block
    long long need = (nvec + threads - 1) / threads;
    if (need < 1) need = 1;
    // 8192 blocks = 65536 wave32s: saturates every WGP while giving each
    // thread ~8 float4s so the uniform-constant preamble is amortized.
    const int blocks = (int)(need < 8192 ? need : 8192);

    dsq_kernel<<<blocks, threads, 0, stream>>>(x, kp, up, out, nvec, n);
}